// LossAF_89541478187420
// MI455X (gfx1250) — compile-verified
//
#include <hip/hip_runtime.h>
#include <hip/hip_bf16.h>

// ---------------- problem constants (match reference) ----------------
#define B_IMG   16
#define S_GRID  80
#define CAND    19200            // 3*80*80
#define NGT     50
#define NCLS    3
#define E_PRED  7                // 4 + NCLS
#define TOPKK   10
#define MAXK    (NGT*TOPKK)      // 500 max unique candidates
#define STRIDE_PX 8.0f           // 640/80
#define EPSP    1e-6f
#define CEPS    1e-7f
#define BIG_T   1.0e30f

typedef float v2f __attribute__((ext_vector_type(2)));
typedef float v8f __attribute__((ext_vector_type(8)));

__device__ __constant__ float d_CW[NCLS] = {
    1.1757211179195934f, 0.09527723808100434f, 1.7290016439994023f };

__device__ __forceinline__ float sigm(float x)  { return 1.0f / (1.0f + expf(-x)); }
__device__ __forceinline__ float softp(float x) { return fmaxf(x, 0.0f) + log1pf(expf(-fabsf(x))); }

// =====================================================================
// Kernel 1: decode anchors, build WMMA A-rows, probs, background focal
// grid = B*CAND/256, block = 256 (one image per block: 19200 % 256 == 0)
// =====================================================================
__global__ __launch_bounds__(256)
void k_decode(const float* __restrict__ preds, float* __restrict__ dec,
              float* __restrict__ amat, float* __restrict__ prob,
              float* __restrict__ bg)
{
    const int gid = blockIdx.x * 256 + threadIdx.x;      // [0, B*CAND)
    const int b   = gid / CAND;
    const int j   = gid % CAND;
    const int cell = j % (S_GRID * S_GRID);
    const float gy = (float)(cell / S_GRID);
    const float gx = (float)(cell % S_GRID);

    const float* p = preds + (size_t)gid * E_PRED;
    const float px = (sigm(p[0]) * 2.0f - 0.5f + gx) * STRIDE_PX;
    const float py = (sigm(p[1]) * 2.0f - 0.5f + gy) * STRIDE_PX;
    const float pw = softp(p[2]) * STRIDE_PX;
    const float ph = softp(p[3]) * STRIDE_PX;

    float* d = dec + (size_t)gid * 4;
    d[0] = px - 0.5f * pw;  d[1] = py - 0.5f * ph;
    d[2] = px + 0.5f * pw;  d[3] = py + 0.5f * ph;

    float* a = amat + (size_t)gid * 4;                    // WMMA A row, K=4
    a[0] = px; a[1] = py; a[2] = px * px + py * py; a[3] = 1.0f;

    // background focal: target = 0 for every (anchor, class)
    float fsum = 0.0f;
    #pragma unroll
    for (int c = 0; c < NCLS; ++c) {
        const float l  = p[4 + c];
        const float pr = sigm(l);
        prob[(size_t)gid * NCLS + c] = pr;
        const float ce0 = fmaxf(l, 0.0f) + log1pf(expf(-fabsf(l)));
        fsum += 0.75f * ce0 * pr * pr * d_CW[c];          // alpha_t=1-0.25, (1-p_t)^2=p^2
    }

    __shared__ float red[256];
    red[threadIdx.x] = fsum;
    __syncthreads();
    for (int s = 128; s > 0; s >>= 1) {
        if (threadIdx.x < s) red[threadIdx.x] += red[threadIdx.x + s];
        __syncthreads();
    }
    if (threadIdx.x == 0) atomicAdd(&bg[b], red[0]);
}

// =====================================================================
// Kernel 2: distance matrix via V_WMMA_F32_16X16X4_F32 + exact top-10
// grid = B * 4 (GT column tiles of 16), block = 256 (8 waves split rows)
// dist[a,g] = [px,py,px^2+py^2,1] . [-2cx,-2cy,1,cx^2+cy^2]
// Software-pipelined: A-tile load for t+1 issues before tile t's top-k
// processing, so the insertion VALU work hides the VMEM latency.
// =====================================================================
__global__ __launch_bounds__(256)
void k_topk(const float* __restrict__ amat, const float* __restrict__ gt_boxes,
            int* __restrict__ cidx)
{
    const int b    = blockIdx.x >> 2;
    const int gt0  = (blockIdx.x & 3) * 16;
    const int tid  = threadIdx.x;
    const int lane = tid & 31;
    const int wave = tid >> 5;

    __shared__ float sd[8][32][TOPKK];
    __shared__ int   si[8][32][TOPKK];

    // B operand: column n = lane&15 -> GT gt0+n; resident for whole sweep.
    // lanes 0-15 hold rows K=0 (v.x) and K=1 (v.y); lanes 16-31 hold K=2,K=3.
    const int n = lane & 15;
    const int g = gt0 + n;
    float tcx = 1.0e8f, tcy = 1.0e8f;                     // pad columns: huge dist
    if (g < NGT) {
        const float* gp = gt_boxes + ((size_t)b * NGT + g) * 4;
        tcx = 0.5f * (gp[0] + gp[2]);
        tcy = 0.5f * (gp[1] + gp[3]);
    }
    v2f bop;
    if (lane < 16) { bop.x = -2.0f * tcx;  bop.y = -2.0f * tcy; }
    else           { bop.x = 1.0f;         bop.y = tcx * tcx + tcy * tcy; }

    // per-lane top-10 (register resident, fully unrolled dynamic replace)
    float tv[TOPKK]; int ti[TOPKK];
    #pragma unroll
    for (int k = 0; k < TOPKK; ++k) { tv[k] = BIG_T; ti[k] = 0; }
    float worst = BIG_T;

    const int tilesPerWave = (CAND / 16) / 8;             // 150
    const int rowOff  = (lane < 16) ? 0 : 8;
    const int laneOff = (lane < 16) ? 0 : 2;              // A half: K=0,1 vs K=2,3

    // per-lane base pointer into this wave's A-row slice
    const float* abase = amat + ((size_t)b * CAND + (size_t)wave * tilesPerWave * 16
                                 + (lane & 15)) * 4 + laneOff;
    // stride between consecutive tiles for this lane: 16 anchors * 4 floats
    const int astride = 16 * 4;

    // ---- software pipeline: prologue load for tile 0 ----
    v2f aop = *(const v2f*)abase;

    for (int t = 0; t < tilesPerWave; ++t) {
        const v2f cur = aop;
        // issue next tile's load before consuming the current one
        if (t + 1 < tilesPerWave)
            aop = *(const v2f*)(abase + (size_t)(t + 1) * astride);
        // warm the stream a few tiles ahead (global_prefetch_b8)
        if (t + 8 < tilesPerWave)
            __builtin_prefetch(abase + (size_t)(t + 8) * astride, 0, 1);

        v8f cz = {0.f,0.f,0.f,0.f,0.f,0.f,0.f,0.f};
        v8f dd = __builtin_amdgcn_wmma_f32_16x16x4_f32(
                     false, cur, false, bop, (short)0, cz, false, false);

        const int tileBase = (wave * tilesPerWave + t) * 16;
        #pragma unroll
        for (int r = 0; r < 8; ++r) {
            const float dval = dd[r];                     // row M = r (+8 hi half)
            const int   didx = tileBase + rowOff + r;
            if (dval < worst) {
                int wi = 0; float wv = tv[0];
                #pragma unroll
                for (int k = 1; k < TOPKK; ++k) if (tv[k] > wv) { wv = tv[k]; wi = k; }
                #pragma unroll
                for (int k = 0; k < TOPKK; ++k) if (k == wi) { tv[k] = dval; ti[k] = didx; }
                worst = tv[0];
                #pragma unroll
                for (int k = 1; k < TOPKK; ++k) worst = fmaxf(worst, tv[k]);
            }
        }
    }

    #pragma unroll
    for (int k = 0; k < TOPKK; ++k) { sd[wave][lane][k] = tv[k]; si[wave][lane][k] = ti[k]; }
    __syncthreads();

    // merge: column n served by lanes {n, n+16} of each of the 8 waves (160 cands)
    if (tid < 16) {
        const int gg = gt0 + tid;
        if (gg < NGT) {
            for (int r = 0; r < TOPKK; ++r) {
                float best = 2.0f * BIG_T; int bw = 0, bl = 0, bk = 0, bidx = 0;
                for (int w = 0; w < 8; ++w)
                    for (int h = 0; h < 2; ++h) {
                        const int l2 = tid + h * 16;
                        for (int k = 0; k < TOPKK; ++k) {
                            const float v = sd[w][l2][k];
                            if (v < best) { best = v; bw = w; bl = l2; bk = k; bidx = si[w][l2][k]; }
                        }
                    }
                sd[bw][bl][bk] = 2.0f * BIG_T;            // consume
                cidx[((size_t)b * NGT + gg) * TOPKK + r] = bidx;
            }
        }
    }
}

// =====================================================================
// Kernel 3: per-image dedupe -> compact cost matrix -> 50-round greedy
//           -> CIoU box loss + focal class corrections
// grid = B, block = 256
// =====================================================================
__global__ __launch_bounds__(256)
void k_match(const float* __restrict__ preds, const float* __restrict__ dec,
             const float* __restrict__ prob, const float* __restrict__ gt_boxes,
             const int* __restrict__ gt_labels, const int* __restrict__ cidx,
             float* __restrict__ costb, float* __restrict__ lbArr,
             float* __restrict__ corrArr, int* __restrict__ nposArr)
{
    const int b = blockIdx.x, tid = threadIdx.x;
    __shared__ unsigned mask[(CAND + 31) / 32];           // 600 words
    __shared__ int   uniq[MAXK];
    __shared__ int   nUniq;
    __shared__ int   rowUsed[MAXK];
    __shared__ int   colUsed[NGT];
    __shared__ int   anchorG[NGT];
    __shared__ float gtb[NGT][4];
    __shared__ int   gtl[NGT];
    __shared__ float rv[256]; __shared__ int ri[256];
    __shared__ float accLB, accCorr; __shared__ int accN;

    for (int i = tid; i < (CAND + 31) / 32; i += 256) mask[i] = 0u;
    for (int i = tid; i < MAXK; i += 256) rowUsed[i] = 0;
    for (int i = tid; i < NGT;  i += 256) { colUsed[i] = 0; anchorG[i] = -1; }
    if (tid == 0) { nUniq = 0; accLB = 0.f; accCorr = 0.f; accN = 0; }
    for (int i = tid; i < NGT; i += 256) {
        const float* gp = gt_boxes + ((size_t)b * NGT + i) * 4;
        gtb[i][0] = gp[0]; gtb[i][1] = gp[1]; gtb[i][2] = gp[2]; gtb[i][3] = gp[3];
        gtl[i] = gt_labels[b * NGT + i];
    }
    __syncthreads();

    // dedupe candidate union via LDS bitmask
    for (int i = tid; i < NGT * TOPKK; i += 256) {
        const int idx = cidx[(size_t)b * NGT * TOPKK + i];
        const unsigned bit = 1u << (idx & 31);
        const unsigned old = atomicOr(&mask[idx >> 5], bit);
        if (!(old & bit)) { const int pos = atomicAdd(&nUniq, 1); uniq[pos] = idx; }
    }
    __syncthreads();
    const int K = nUniq;

    // compact cost matrix [K x NGT]
    float* cb = costb + (size_t)b * MAXK * NGT;
    for (int e = tid; e < K * NGT; e += 256) {
        const int k = e / NGT, g = e % NGT, a = uniq[k];
        const float* d = dec + ((size_t)b * CAND + a) * 4;
        const float px1=d[0], py1=d[1], px2=d[2], py2=d[3];
        const float tx1=gtb[g][0], ty1=gtb[g][1], tx2=gtb[g][2], ty2=gtb[g][3];
        const float a1  = fmaxf(px2-px1,0.f)*fmaxf(py2-py1,0.f);
        const float a2  = fmaxf(tx2-tx1,0.f)*fmaxf(ty2-ty1,0.f);
        const float inter = fmaxf(fminf(px2,tx2)-fmaxf(px1,tx1),0.f) *
                            fmaxf(fminf(py2,ty2)-fmaxf(py1,ty1),0.f);
        const float iou = fminf(fmaxf(inter / (a1 + a2 - inter + CEPS), 0.f), 1.f);
        float pr = prob[((size_t)b * CAND + a) * NCLS + gtl[g]];
        pr = fminf(fmaxf(pr, EPSP), 1.0f - EPSP);
        cb[e] = 0.5f * (-logf(pr)) + 6.0f * (1.0f - iou);
    }
    __syncthreads();

    // greedy one-to-one: 50 rounds of global argmin with consumed flags
    for (int round = 0; round < NGT; ++round) {
        float lm = BIG_T; int li = -1;
        for (int e = tid; e < K * NGT; e += 256) {
            const int k = e / NGT, g = e % NGT;
            if (rowUsed[k] | colUsed[g]) continue;
            const float v = cb[e];
            if (v < lm) { lm = v; li = e; }
        }
        rv[tid] = lm; ri[tid] = li;
        __syncthreads();
        for (int s = 128; s > 0; s >>= 1) {
            if (tid < s && rv[tid + s] < rv[tid]) { rv[tid] = rv[tid + s]; ri[tid] = ri[tid + s]; }
            __syncthreads();
        }
        if (tid == 0 && ri[0] >= 0 && rv[0] < 0.5f * BIG_T) {
            const int k = ri[0] / NGT, g = ri[0] % NGT;
            rowUsed[k] = 1; colUsed[g] = 1; anchorG[g] = uniq[k];
        }
        __syncthreads();
    }

    // matched losses: CIoU + focal correction (target flips 0 -> 1 at match)
    if (tid < NGT) {
        const int g = tid, a = anchorG[g];
        if (a >= 0) {
            const float* d = dec + ((size_t)b * CAND + a) * 4;
            const float px1=d[0], py1=d[1], px2=d[2], py2=d[3];
            const float tx1=gtb[g][0], ty1=gtb[g][1], tx2=gtb[g][2], ty2=gtb[g][3];
            const float pw=fmaxf(px2-px1,CEPS), ph=fmaxf(py2-py1,CEPS);
            const float tw=fmaxf(tx2-tx1,CEPS), th=fmaxf(ty2-ty1,CEPS);
            const float inter = fmaxf(fminf(px2,tx2)-fmaxf(px1,tx1),0.f) *
                                fmaxf(fminf(py2,ty2)-fmaxf(py1,ty1),0.f);
            const float uni = pw*ph + tw*th - inter + CEPS;
            const float iou = inter / uni;
            const float cw = fmaxf(px2,tx2)-fminf(px1,tx1);
            const float ch = fmaxf(py2,ty2)-fminf(py1,ty1);
            const float c2 = cw*cw + ch*ch + CEPS;
            const float dx = px1+px2-tx1-tx2, dy = py1+py2-ty1-ty2;
            const float rho2 = (dx*dx + dy*dy) * 0.25f;
            const float dat = atanf(tw/th) - atanf(pw/ph);
            const float v = 0.4052847345693511f * dat * dat;   // 4/pi^2
            const float alpha = v / (v - iou + 1.0f + CEPS);
            atomicAdd(&accLB, 1.0f - (iou - rho2 / c2 - alpha * v));
            atomicAdd(&accN, 1);

            const int lbl = gtl[g];
            const float l  = preds[((size_t)b * CAND + a) * E_PRED + 4 + lbl];
            const float pr  = sigm(l);
            const float lg  = log1pf(expf(-fabsf(l)));
            const float ce1 = fmaxf(l, 0.f) - l + lg;
            const float ce0 = fmaxf(l, 0.f) + lg;
            const float f1  = 0.25f * ce1 * (1.f - pr) * (1.f - pr);
            const float f0  = 0.75f * ce0 * pr * pr;
            atomicAdd(&accCorr, (f1 - f0) * d_CW[lbl]);
        }
    }
    __syncthreads();
    if (tid == 0) { lbArr[b] = accLB; corrArr[b] = accCorr; nposArr[b] = accN; }
}

// =====================================================================
// Kernel 4: combine scalars
// =====================================================================
__global__ void k_final(const float* __restrict__ bg, const float* __restrict__ lbArr,
                        const float* __restrict__ corrArr, const int* __restrict__ nposArr,
                        float* __restrict__ out)
{
    if (threadIdx.x == 0) {
        float LB = 0.f, LC = 0.f; int NP = 0;
        for (int b = 0; b < B_IMG; ++b) { LB += lbArr[b]; LC += bg[b] + corrArr[b]; NP += nposArr[b]; }
        const float norm = fmaxf(1.0f, (float)NP);
        out[0] = 7.5f * LB / norm + 0.5f * LC / norm;
    }
}

// =====================================================================
// launch
// =====================================================================
extern "C" void kernel_launch(void* const* d_in, const int* in_sizes, int n_in,
                              void* d_out, int out_size, void* d_ws, size_t ws_size,
                              hipStream_t stream)
{
    const float* preds     = (const float*)d_in[0];   // [16,3,80,80,7]
    const float* gt_boxes  = (const float*)d_in[1];   // [16,50,4]
    const int*   gt_labels = (const int*)d_in[2];     // [16,50]
    float*       out       = (float*)d_out;

    // workspace carve (~15.1 MB)
    float* dec   = (float*)d_ws;                          // B*CAND*4
    float* amat  = dec   + (size_t)B_IMG * CAND * 4;      // B*CAND*4
    float* prob  = amat  + (size_t)B_IMG * CAND * 4;      // B*CAND*3
    int*   cidx  = (int*)(prob + (size_t)B_IMG * CAND * NCLS);     // B*50*10
    float* costb = (float*)(cidx + (size_t)B_IMG * NGT * TOPKK);   // B*500*50
    float* bg    = costb + (size_t)B_IMG * MAXK * NGT;    // B
    float* lbA   = bg    + B_IMG;                         // B
    float* corrA = lbA   + B_IMG;                         // B
    int*   nposA = (int*)(corrA + B_IMG);                 // B

    hipMemsetAsync(bg, 0, B_IMG * sizeof(float), stream);

    k_decode<<<(B_IMG * CAND) / 256, 256, 0, stream>>>(preds, dec, amat, prob, bg);
    k_topk  <<<B_IMG * 4, 256, 0, stream>>>(amat, gt_boxes, cidx);
    k_match <<<B_IMG, 256, 0, stream>>>(preds, dec, prob, gt_boxes, gt_labels,
                                        cidx, costb, lbA, corrA, nposA);
    k_final <<<1, 32, 0, stream>>>(bg, lbA, corrA, nposA, out);
}